// GraphCrossAttenNet_52553219833885
// MI455X (gfx1250) — compile-verified
//
#include <hip/hip_runtime.h>
#include <hip/hip_bf16.h>
#include <float.h>
#include <stdint.h>

// Problem dims (fixed by the reference)
#define N_NODES 32768
#define N_EDGES 131072
#define NH      4
#define RNA     128
#define PROT    32
#define F0      64
#define CIN     96     // F0 + PROT
#define MAX_K   160    // largest GEMM K (RNA + PROT)

typedef float v2f __attribute__((ext_vector_type(2)));
typedef float v8f __attribute__((ext_vector_type(8)));

// gfx1250 async copy (global -> LDS, ASYNCcnt-tracked). Guarded so the file
// compiles on toolchains that do not expose the builtins. Probe round 2 showed
// the builtin params are int pointees: (as1 int*, as3 int*, imm off, imm cpol).
#if __has_builtin(__builtin_amdgcn_global_load_async_to_lds_b32) && \
    __has_builtin(__builtin_amdgcn_s_wait_asynccnt)
#define USE_ASYNC_LDS 1
typedef __attribute__((address_space(1))) int* as1_int_ptr;
typedef __attribute__((address_space(3))) int* as3_int_ptr;
#else
#define USE_ASYNC_LDS 0
#endif

// ---------------------------------------------------------------------------
// Generic f32 WMMA GEMM:  C[M,N] = concat(A1[M,K1], A2[M,K-K1]) * B[K,N] + bias
// One wave computes one 16x16 C tile via V_WMMA_F32_16X16X4_F32 (K stepped 4).
// Block = 128 threads = 4 waves tiling M; grid = (M/64, N/16).
// The K x 16 B panel is staged once per block into LDS (async copy when
// available), shared by all 4 waves. All dims are exact multiples -> every
// wave fully active (EXEC all-ones, required for WMMA).
// ---------------------------------------------------------------------------
__global__ void gemm_f32_wmma(const float* __restrict__ A1,
                              const float* __restrict__ A2, int K1, int K,
                              const float* __restrict__ B,
                              const float* __restrict__ bias,
                              float* __restrict__ C, int N) {
  __shared__ float Bs[MAX_K * 16];

  const int lane = threadIdx.x & 31;
  const int wave = threadIdx.x >> 5;
  const int mt   = (blockIdx.x * 4 + wave) * 16;   // M tile origin
  const int nt   = blockIdx.y * 16;                // N tile origin
  const int half = lane >> 4;                      // 0: lanes 0-15, 1: 16-31
  const int lr   = lane & 15;
  const int K2   = K - K1;

  // ---- Stage B panel (K x 16) into LDS, cooperatively across the block ----
  for (int idx = threadIdx.x; idx < K * 16; idx += 128) {
    const int kk = idx >> 4;
    const int nn = idx & 15;
    const float* gsrc = B + (size_t)kk * N + nt + nn;
#if USE_ASYNC_LDS
    __builtin_amdgcn_global_load_async_to_lds_b32(
        (as1_int_ptr)(uintptr_t)gsrc, (as3_int_ptr)(uintptr_t)(Bs + idx), 0, 0);
#else
    Bs[idx] = *gsrc;
#endif
  }
#if USE_ASYNC_LDS
  __builtin_amdgcn_s_wait_asynccnt(0);   // drain this wave's async copies
#endif
  __syncthreads();                        // publish LDS to all 4 waves

  // ---- WMMA main loop (A fragment: lanes 0-15 K=k..k+1, 16-31 K=k+2..3) ---
  v8f acc = {0.f, 0.f, 0.f, 0.f, 0.f, 0.f, 0.f, 0.f};

  {
    const float* ap = A1 + (size_t)(mt + lr) * K1 + half * 2;
    const float* bp = Bs + half * 2 * 16 + lr;
    for (int k = 0; k < K1; k += 4) {
      const v2f a = *(const v2f*)ap;      // 8B-aligned -> global_load_b64
      v2f b;
      b[0] = bp[0];                        // B[ka,   n] from LDS
      b[1] = bp[16];                       // B[ka+1, n] from LDS
      acc = __builtin_amdgcn_wmma_f32_16x16x4_f32(false, a, false, b,
                                                  (short)0, acc, false, false);
      ap += 4;
      bp += 64;
    }
  }
  if (K2 > 0) {                            // concat() tail (prot features)
    const float* ap = A2 + (size_t)(mt + lr) * K2 + half * 2;
    const float* bp = Bs + (size_t)(K1 + half * 2) * 16 + lr;
    for (int k = 0; k < K2; k += 4) {
      const v2f a = *(const v2f*)ap;
      v2f b;
      b[0] = bp[0];
      b[1] = bp[16];
      acc = __builtin_amdgcn_wmma_f32_16x16x4_f32(false, a, false, b,
                                                  (short)0, acc, false, false);
      ap += 4;
      bp += 64;
    }
  }

  // C/D layout: VGPR j = row (mt + half*8 + j), col (nt + lr)
  const int   col = nt + lr;
  const float bv  = bias ? bias[col] : 0.f;
#pragma unroll
  for (int j = 0; j < 8; ++j) {
    const int row = mt + half * 8 + j;
    C[(size_t)row * N + col] = acc[j] + bv;
  }
}

// ---------------------------------------------------------------------------
// s_src[n,h] = <proj[n,h,:], a_src[h,:]>, s_trg likewise. One thread per (n,h).
// ---------------------------------------------------------------------------
__global__ void node_scores(const float* __restrict__ proj,
                            const float* __restrict__ a_src,
                            const float* __restrict__ a_trg,
                            float* __restrict__ s_src,
                            float* __restrict__ s_trg, int F) {
  const int idx = blockIdx.x * blockDim.x + threadIdx.x;  // n*NH + h
  if (idx >= N_NODES * NH) return;
  const int h = idx & (NH - 1);
  const float* p = proj + (size_t)idx * F;
  float ss = 0.f, st = 0.f;
  for (int f = 0; f < F; ++f) {
    const float v = p[f];
    ss += v * a_src[h * F + f];
    st += v * a_trg[h * F + f];
  }
  s_src[idx] = ss;
  s_trg[idx] = st;
}

// ---------------------------------------------------------------------------
// Per-edge leaky-relu scores + per-block max (pass 1 of global max).
// One thread per edge, E/256 = 512 blocks exactly.
// ---------------------------------------------------------------------------
__global__ void edge_scores(const int* __restrict__ src,
                            const int* __restrict__ trg,
                            const float* __restrict__ s_src,
                            const float* __restrict__ s_trg,
                            float* __restrict__ scores,
                            float* __restrict__ pmax) {
  __shared__ float smax[256];
  const int e = blockIdx.x * 256 + threadIdx.x;
  const int s = src[e], t = trg[e];
  float local = -FLT_MAX;
#pragma unroll
  for (int h = 0; h < NH; ++h) {
    float sc = s_src[s * NH + h] + s_trg[t * NH + h];
    sc = sc > 0.f ? sc : 0.2f * sc;                 // leaky_relu(0.2)
    scores[e * NH + h] = sc;
    local = fmaxf(local, sc);
  }
  smax[threadIdx.x] = local;
  __syncthreads();
  for (int w = 128; w > 0; w >>= 1) {
    if (threadIdx.x < w)
      smax[threadIdx.x] = fmaxf(smax[threadIdx.x], smax[threadIdx.x + w]);
    __syncthreads();
  }
  if (threadIdx.x == 0) pmax[blockIdx.x] = smax[0];
}

// Pass 2: single-block reduce of the partial maxima (deterministic).
__global__ void reduce_max(const float* __restrict__ pmax, int n,
                           float* __restrict__ gmax) {
  __shared__ float smax[256];
  float local = -FLT_MAX;
  for (int i = threadIdx.x; i < n; i += 256) local = fmaxf(local, pmax[i]);
  smax[threadIdx.x] = local;
  __syncthreads();
  for (int w = 128; w > 0; w >>= 1) {
    if (threadIdx.x < w)
      smax[threadIdx.x] = fmaxf(smax[threadIdx.x], smax[threadIdx.x + w]);
    __syncthreads();
  }
  if (threadIdx.x == 0) gmax[0] = smax[0];
}

// ---------------------------------------------------------------------------
// ex = exp(score - gmax) (in place), denom[grp,h] += ex. grp = trg (GAT) or
// src (cross). One thread per (e,h).
// ---------------------------------------------------------------------------
__global__ void edge_exp_denom(const int* __restrict__ grp,
                               float* __restrict__ scores,  // in: score, out: ex
                               const float* __restrict__ gmax,
                               float* __restrict__ denom) {
  const int i = blockIdx.x * blockDim.x + threadIdx.x;      // e*NH + h
  if (i >= N_EDGES * NH) return;
  const int h = i & (NH - 1);
  const int e = i >> 2;
  const float ex = __expf(scores[i] - gmax[0]);
  scores[i] = ex;
  atomicAdd(&denom[grp[e] * NH + h], ex);
}

// ---------------------------------------------------------------------------
// out_acc[trg,f] += proj[src,h,f] * attn(e,h) / NH   (head-mean folded in).
// One thread per (e,h,f).
// ---------------------------------------------------------------------------
__global__ void edge_aggregate(const int* __restrict__ src,
                               const int* __restrict__ trg,
                               const int* __restrict__ grp,
                               const float* __restrict__ ex,
                               const float* __restrict__ denom,
                               const float* __restrict__ proj,
                               float* __restrict__ out_acc, int F) {
  const long long i = (long long)blockIdx.x * blockDim.x + threadIdx.x;
  if (i >= (long long)N_EDGES * NH * F) return;
  const int f  = (int)(i % F);
  const int eh = (int)(i / F);
  const int h  = eh & (NH - 1);
  const int e  = eh >> 2;
  const float attn = ex[eh] / (denom[grp[e] * NH + h] + 1e-16f);
  const float v = proj[((size_t)src[e] * NH + h) * F + f] * attn * (1.0f / NH);
  atomicAdd(&out_acc[(size_t)trg[e] * F + f], v);
}

// out[n,f] += x[n,f] + bias[f]   (skip connection + bias after head-mean)
__global__ void finalize_layer(float* __restrict__ out_acc,
                               const float* __restrict__ x,
                               const float* __restrict__ bias, int F) {
  const int i = blockIdx.x * blockDim.x + threadIdx.x;
  if (i >= N_NODES * F) return;
  out_acc[i] += x[i] + bias[i % F];
}

// ---------------------------------------------------------------------------
extern "C" void kernel_launch(void* const* d_in, const int* in_sizes, int n_in,
                              void* d_out, int out_size, void* d_ws,
                              size_t ws_size, hipStream_t stream) {
  const float* rna    = (const float*)d_in[0];
  const float* prot   = (const float*)d_in[1];
  const int*   eidx   = (const int*)d_in[2];
  const float* Wgp    = (const float*)d_in[3];
  const float* bgp    = (const float*)d_in[4];
  const float* Wgr    = (const float*)d_in[5];
  const float* bgr    = (const float*)d_in[6];
  const float* Wg     = (const float*)d_in[7];
  const float* ag_src = (const float*)d_in[8];
  const float* ag_trg = (const float*)d_in[9];
  const float* bg     = (const float*)d_in[10];
  const float* Wcp    = (const float*)d_in[11];
  const float* bcp    = (const float*)d_in[12];
  const float* Wcr    = (const float*)d_in[13];
  const float* bcr    = (const float*)d_in[14];
  const float* Wc     = (const float*)d_in[15];
  const float* ac_src = (const float*)d_in[16];
  const float* ac_trg = (const float*)d_in[17];
  const float* bc     = (const float*)d_in[18];

  const int* src = eidx;            // edge_index[0]
  const int* trg = eidx + N_EDGES;  // edge_index[1]

  float* gat_final   = (float*)d_out;                          // [N, RNA]
  float* cross_final = (float*)d_out + (size_t)N_NODES * RNA;  // [N, RNA+PROT]

  // Workspace carve-out (~133 MB of f32)
  float* ws = (float*)d_ws;
  size_t off = 0;
  auto alloc = [&](size_t n) { float* p = ws + off; off += n; return p; };
  float* rna_p     = alloc((size_t)N_NODES * F0);
  float* cross_p   = alloc((size_t)N_NODES * CIN);
  float* proj_g    = alloc((size_t)N_NODES * NH * F0);
  float* proj_c    = alloc((size_t)N_NODES * NH * CIN);
  float* ssrc_g    = alloc((size_t)N_NODES * NH);
  float* strg_g    = alloc((size_t)N_NODES * NH);
  float* ssrc_c    = alloc((size_t)N_NODES * NH);
  float* strg_c    = alloc((size_t)N_NODES * NH);
  float* sc_g      = alloc((size_t)N_EDGES * NH);   // scores -> ex (in place)
  float* sc_c      = alloc((size_t)N_EDGES * NH);
  float* denom_g   = alloc((size_t)N_NODES * NH);
  float* denom_c   = alloc((size_t)N_NODES * NH);
  float* gat_out   = alloc((size_t)N_NODES * F0);
  float* cross_out = alloc((size_t)N_NODES * CIN);
  float* pmax_g    = alloc(512);
  float* pmax_c    = alloc(512);
  float* gmax_g    = alloc(1);
  float* gmax_c    = alloc(1);

  // Zero the accumulators every call (graph-capture safe).
  (void)hipMemsetAsync(denom_g, 0, sizeof(float) * N_NODES * NH, stream);
  (void)hipMemsetAsync(denom_c, 0, sizeof(float) * N_NODES * NH, stream);
  (void)hipMemsetAsync(gat_out, 0, sizeof(float) * (size_t)N_NODES * F0, stream);
  (void)hipMemsetAsync(cross_out, 0, sizeof(float) * (size_t)N_NODES * CIN,
                       stream);

  const dim3 gblk(128);                  // 4 waves / block
  auto ggrid = [](int n) { return dim3(N_NODES / 64, n / 16); };

  // Input projections (cross path reads rna|prot split, never materialized)
  gemm_f32_wmma<<<ggrid(F0), gblk, 0, stream>>>(rna, nullptr, RNA, RNA, Wgp,
                                                bgp, rna_p, F0);
  gemm_f32_wmma<<<ggrid(CIN), gblk, 0, stream>>>(rna, prot, RNA, RNA + PROT,
                                                 Wcp, bcp, cross_p, CIN);
  // Head lifts: proj = x @ W  ([N, NH*F])
  gemm_f32_wmma<<<ggrid(NH * F0), gblk, 0, stream>>>(rna_p, nullptr, F0, F0,
                                                     Wg, nullptr, proj_g,
                                                     NH * F0);
  gemm_f32_wmma<<<ggrid(NH * CIN), gblk, 0, stream>>>(cross_p, nullptr, CIN,
                                                      CIN, Wc, nullptr, proj_c,
                                                      NH * CIN);

  // Node attention scores
  node_scores<<<N_NODES * NH / 256, 256, 0, stream>>>(proj_g, ag_src, ag_trg,
                                                      ssrc_g, strg_g, F0);
  node_scores<<<N_NODES * NH / 256, 256, 0, stream>>>(proj_c, ac_src, ac_trg,
                                                      ssrc_c, strg_c, CIN);

  // Edge scores + global max (two-pass, deterministic)
  edge_scores<<<N_EDGES / 256, 256, 0, stream>>>(src, trg, ssrc_g, strg_g,
                                                 sc_g, pmax_g);
  edge_scores<<<N_EDGES / 256, 256, 0, stream>>>(src, trg, ssrc_c, strg_c,
                                                 sc_c, pmax_c);
  reduce_max<<<1, 256, 0, stream>>>(pmax_g, N_EDGES / 256, gmax_g);
  reduce_max<<<1, 256, 0, stream>>>(pmax_c, N_EDGES / 256, gmax_c);

  // Softmax numerators + segment denominators (GAT groups by trg, cross by src)
  edge_exp_denom<<<N_EDGES * NH / 256, 256, 0, stream>>>(trg, sc_g, gmax_g,
                                                         denom_g);
  edge_exp_denom<<<N_EDGES * NH / 256, 256, 0, stream>>>(src, sc_c, gmax_c,
                                                         denom_c);

  // Scatter-add aggregation (head-mean 1/NH folded in)
  edge_aggregate<<<(unsigned)((long long)N_EDGES * NH * F0 / 256), 256, 0,
                   stream>>>(src, trg, trg, sc_g, denom_g, proj_g, gat_out, F0);
  edge_aggregate<<<(unsigned)((long long)N_EDGES * NH * CIN / 256), 256, 0,
                   stream>>>(src, trg, src, sc_c, denom_c, proj_c, cross_out,
                             CIN);

  // Skip connection + per-layer bias
  finalize_layer<<<N_NODES * F0 / 256, 256, 0, stream>>>(gat_out, rna_p, bg,
                                                         F0);
  finalize_layer<<<N_NODES * CIN / 256, 256, 0, stream>>>(cross_out, cross_p,
                                                          bc, CIN);

  // Reprojections straight into d_out (cross reshape is layout-only)
  gemm_f32_wmma<<<ggrid(RNA), gblk, 0, stream>>>(gat_out, nullptr, F0, F0, Wgr,
                                                 bgr, gat_final, RNA);
  gemm_f32_wmma<<<ggrid(RNA + PROT), gblk, 0, stream>>>(cross_out, nullptr,
                                                        CIN, CIN, Wcr, bcr,
                                                        cross_final,
                                                        RNA + PROT);
}